// CenterLoss_57389353009307
// MI455X (gfx1250) — compile-verified
//
#include <hip/hip_runtime.h>
#include <hip/hip_bf16.h>

typedef float v2f __attribute__((ext_vector_type(2)));
typedef float v8f __attribute__((ext_vector_type(8)));

#define D_DIM 1024  // feature dimension (reference: D = 1024, divisible by 256*4)

// One block (256 threads = 8 wave32) per sample row.
// Each thread: one float4 of x and of the gathered center row -> squared diff.
// Wave-level reduction via V_WMMA_F32_16X16X4_F32 with B = ones (layout-robust
// row-sum trick), then LDS combine across the 8 waves, one atomicAdd per block.
__global__ __launch_bounds__(256) void center_loss_kernel(
    const float* __restrict__ x,
    const int*   __restrict__ tgt,
    const float* __restrict__ ctr,
    float*       __restrict__ out) {
  const int row  = blockIdx.x;
  const int tid  = threadIdx.x;
  const int lane = tid & 31;
  const int wid  = tid >> 5;

  const int t = tgt[row];  // uniform per block -> scalarized by compiler

  const float4* __restrict__ xr = (const float4*)(x   + (size_t)row * D_DIM);
  const float4* __restrict__ cr = (const float4*)(ctr + (size_t)t   * D_DIM);

  const float4 xa = xr[tid];   // coalesced 16B per lane
  const float4 ca = cr[tid];   // coalesced within the gathered 4KB row

  const float dx = xa.x - ca.x;
  const float dy = xa.y - ca.y;
  const float dz = xa.z - ca.z;
  const float dw = xa.w - ca.w;
  const float s  = dx * dx + dy * dy + dz * dz + dw * dw;

  // ---- wave32 reduction via WMMA -------------------------------------------
  // A (16x4 f32, 2 VGPRs): lane L<16 holds A[L][0] (v0) and A[L][1] (v1);
  // lane L>=16 holds A[L-16][2] and A[L-16][3]. Put the partial in v0, 0 in v1.
  // B = all ones (4x16): invariant to any lane-striping convention.
  // D[r][n] = sum_k A[r][k] = s_r + s_{r+16}  for every n.
  v2f a; a.x = s;    a.y = 0.0f;
  v2f b; b.x = 1.0f; b.y = 1.0f;
  v8f c = {};
  c = __builtin_amdgcn_wmma_f32_16x16x4_f32(
      /*neg_a=*/false, a, /*neg_b=*/false, b,
      /*c_mod=*/(short)0, c, /*reuse_a=*/false, /*reuse_b=*/false);

  // C/D layout: c[v] @ lanes 0-15 -> M=v, @ lanes 16-31 -> M=v+8.
  // So per-lane sum over the 8 accumulators gives:
  //   lanes 0-15 : rows 0..7, lanes 16-31 : rows 8..15.
  float v = c[0] + c[1] + c[2] + c[3] + c[4] + c[5] + c[6] + c[7];
  float wave_total = __shfl(v, 0, 32) + __shfl(v, 16, 32);  // full 32-lane sum

  // ---- cross-wave combine ---------------------------------------------------
  __shared__ float ws[8];
  if (lane == 0) ws[wid] = wave_total;
  __syncthreads();
  if (tid == 0) {
    float blk = 0.0f;
#pragma unroll
    for (int i = 0; i < 8; ++i) blk += ws[i];
    atomicAdd(out, 0.5f * blk);
  }
}

extern "C" void kernel_launch(void* const* d_in, const int* in_sizes, int n_in,
                              void* d_out, int out_size, void* d_ws, size_t ws_size,
                              hipStream_t stream) {
  const float* x   = (const float*)d_in[0];  // [N, D] f32
  const int*   tgt = (const int*)d_in[1];    // [N] int32
  const float* ctr = (const float*)d_in[2];  // [C, D] f32
  float* out = (float*)d_out;                // scalar f32

  const int N = in_sizes[0] / D_DIM;         // 4096

  // Deterministic across graph replays: zero the accumulator every launch.
  hipMemsetAsync(out, 0, sizeof(float), stream);

  center_loss_kernel<<<N, 256, 0, stream>>>(x, tgt, ctr, out);
}